// GAE_Encoder_90718299226206
// MI455X (gfx1250) — compile-verified
//
#include <hip/hip_runtime.h>
#include <math.h>

#define EPSF 1e-15f
#define KT 32           // K-tile depth staged in LDS
#define LDSPAD 65       // row stride (64 + 1) to spread banks

typedef __attribute__((ext_vector_type(2))) float v2f;
typedef __attribute__((ext_vector_type(8))) float v8f;

// ---------------------------------------------------------------------------
// WMMA fp32 GEMM:  C[M,N] = op(A) @ B  (+bias) (optional relu)
//   TA=false: A is [M,K] row-major.   TA=true: A stored [K,M] (C = A^T B).
//   B is [K,N] row-major. Requires M%16==0, N%64==0, K%4==0.
// Block = 128 threads (4 waves). Block computes a 64x64 C tile; wave w owns
// M-rows [m0+16w, m0+16w+16) x all 64 N cols (4 WMMA accumulators).
// A/B K-tiles (depth 32) double-buffered in LDS; B reused by all 4 waves.
// Fragment layouts per CDNA5 ISA 7.12.2:
//   A frag: vgpr v = A[M=lane%16][K = kt + v + 2*(lane/16)]
//   B frag: vgpr v = B[K = kt + v + 2*(lane/16)][N = lane%16 (+16t)]
//   C frag: vgpr v = C[M = v + 8*(lane/16)][N = lane%16 (+16t)]
// ---------------------------------------------------------------------------
template <bool TA>
__global__ __launch_bounds__(128) void wmma_gemm(
    const float* __restrict__ A, const float* __restrict__ B,
    float* __restrict__ C, const float* __restrict__ bias,
    int M, int N, int K, int relu)
{
    __shared__ float At[2][KT][LDSPAD];  // [k][m]
    __shared__ float Bt[2][KT][LDSPAD];  // [k][n]

    const int tid  = threadIdx.x;
    const int wave = tid >> 5;
    const int lane = tid & 31;
    const int lo   = lane & 15;
    const int hi   = lane >> 4;
    const int m0   = blockIdx.y << 6;
    const int n0   = blockIdx.x << 6;
    const int mw   = m0 + (wave << 4);

    v8f acc0 = {}, acc1 = {}, acc2 = {}, acc3 = {};

    const int ntiles = (K + KT - 1) / KT;

    auto stage = [&](int kk, int buf) {
        // ---- B tile: 32 k-rows x 64 n, coalesced float4 rows ----
#pragma unroll
        for (int r = 0; r < 4; ++r) {
            const int k = (tid >> 4) + (r << 3);   // 0..31
            const int n = (tid & 15) << 2;         // 0..60
            const int gk = kk + k;
            float4 v = make_float4(0.f, 0.f, 0.f, 0.f);
            if (gk < K) {
                v = *(const float4*)(B + (size_t)gk * N + n0 + n);
                if (r == 0 && gk + KT < K)  // prefetch one K-tile ahead
                    __builtin_prefetch(B + (size_t)(gk + KT) * N + n0 + n, 0, 1);
            }
            Bt[buf][k][n + 0] = v.x; Bt[buf][k][n + 1] = v.y;
            Bt[buf][k][n + 2] = v.z; Bt[buf][k][n + 3] = v.w;
        }
        // ---- A tile into [k][m] layout ----
        if (TA) {   // A stored [K,M]: coalesced float4 along m
#pragma unroll
            for (int r = 0; r < 4; ++r) {
                const int k = (tid >> 4) + (r << 3);
                const int m = (tid & 15) << 2;
                const int gk = kk + k;
                float4 v = make_float4(0.f, 0.f, 0.f, 0.f);
                if (gk < K && (m0 + m) < M) {
                    v = *(const float4*)(A + (size_t)gk * M + m0 + m);
                    if (r == 0 && gk + KT < K)
                        __builtin_prefetch(A + (size_t)(gk + KT) * M + m0 + m, 0, 1);
                }
                At[buf][k][m + 0] = v.x; At[buf][k][m + 1] = v.y;
                At[buf][k][m + 2] = v.z; At[buf][k][m + 3] = v.w;
            }
        } else {    // A stored [M,K]: coalesced float4 along k, transpose into LDS
#pragma unroll
            for (int r = 0; r < 4; ++r) {
                const int m  = (tid >> 3) + (r << 4); // 0..63
                const int k4 = (tid & 7) << 2;        // 0..28
                const int gm = m0 + m;
                float4 v = make_float4(0.f, 0.f, 0.f, 0.f);
                if (gm < M && (kk + k4) < K) {
                    v = *(const float4*)(A + (size_t)gm * K + kk + k4);
                    if (r == 0 && kk + k4 + KT < K)
                        __builtin_prefetch(A + (size_t)gm * K + kk + k4 + KT, 0, 1);
                }
                At[buf][k4 + 0][m] = v.x; At[buf][k4 + 1][m] = v.y;
                At[buf][k4 + 2][m] = v.z; At[buf][k4 + 3][m] = v.w;
            }
        }
    };

    stage(0, 0);
    __syncthreads();

    for (int t = 0; t < ntiles; ++t) {
        const int buf = t & 1;
        if (t + 1 < ntiles) stage((t + 1) * KT, buf ^ 1);

#pragma unroll
        for (int kt = 0; kt < KT; kt += 4) {
            v2f a;
            a.x = At[buf][kt + 2 * hi + 0][(wave << 4) + lo];
            a.y = At[buf][kt + 2 * hi + 1][(wave << 4) + lo];
            v2f b0, b1, b2, b3;
            b0.x = Bt[buf][kt + 2 * hi + 0][lo +  0];
            b0.y = Bt[buf][kt + 2 * hi + 1][lo +  0];
            b1.x = Bt[buf][kt + 2 * hi + 0][lo + 16];
            b1.y = Bt[buf][kt + 2 * hi + 1][lo + 16];
            b2.x = Bt[buf][kt + 2 * hi + 0][lo + 32];
            b2.y = Bt[buf][kt + 2 * hi + 1][lo + 32];
            b3.x = Bt[buf][kt + 2 * hi + 0][lo + 48];
            b3.y = Bt[buf][kt + 2 * hi + 1][lo + 48];
            acc0 = __builtin_amdgcn_wmma_f32_16x16x4_f32(false, a, false, b0, (short)0, acc0, false, false);
            acc1 = __builtin_amdgcn_wmma_f32_16x16x4_f32(false, a, false, b1, (short)0, acc1, false, false);
            acc2 = __builtin_amdgcn_wmma_f32_16x16x4_f32(false, a, false, b2, (short)0, acc2, false, false);
            acc3 = __builtin_amdgcn_wmma_f32_16x16x4_f32(false, a, false, b3, (short)0, acc3, false, false);
        }
        __syncthreads();
    }

    v8f accs[4] = {acc0, acc1, acc2, acc3};
#pragma unroll
    for (int t = 0; t < 4; ++t) {
        const int n = n0 + t * 16 + lo;
        const float bv = bias ? bias[n] : 0.0f;
#pragma unroll
        for (int v = 0; v < 8; ++v) {
            const int m = mw + v + 8 * hi;
            if (m < M) {
                float val = accs[t][v] + bv;
                if (relu) val = fmaxf(val, 0.0f);
                C[(size_t)m * N + n] = val;
            }
        }
    }
}

// ---------------------------------------------------------------------------
// Elementwise / scatter / reduction helpers
// ---------------------------------------------------------------------------
__global__ void fill_kernel(float* __restrict__ p, float v, size_t n) {
    size_t i = (size_t)blockIdx.x * blockDim.x + threadIdx.x;
    if (i < n) p[i] = v;
}

__global__ void edge_deg_kernel(const int* __restrict__ row, const int* __restrict__ col,
                                const float* __restrict__ w,
                                float* __restrict__ deg, float* __restrict__ drow, int E) {
    int e = blockIdx.x * blockDim.x + threadIdx.x;
    if (e < E) {
        atomicAdd(&deg[col[e]], w[e]);
        atomicAdd(&drow[row[e]], w[e]);
    }
}

__global__ void rsqrt_kernel(const float* __restrict__ deg, float* __restrict__ dis, int n) {
    int i = blockIdx.x * blockDim.x + threadIdx.x;
    if (i < n) { float d = deg[i]; dis[i] = d > 0.0f ? rsqrtf(d) : 0.0f; }
}

// acc[i,f] = dis[i]^2 * h[i,f]  (self-loop term of GCN aggregation)
__global__ void selfloop_init_kernel(const float* __restrict__ h, const float* __restrict__ dis,
                                     float* __restrict__ acc, size_t n, int F) {
    size_t idx = (size_t)blockIdx.x * blockDim.x + threadIdx.x;
    if (idx < n) {
        int i = (int)(idx / F);
        float d = dis[i];
        acc[idx] = d * d * h[idx];
    }
}

// acc[col,f] += dis[row]*w*dis[col] * h[row,f]   (F = 512, 4 floats/thread)
__global__ void scatter_gcn_kernel(const int* __restrict__ row, const int* __restrict__ col,
                                   const float* __restrict__ w, const float* __restrict__ dis,
                                   const float* __restrict__ h, float* __restrict__ acc, int E) {
    size_t tid = (size_t)blockIdx.x * blockDim.x + threadIdx.x;
    if (tid >= (size_t)E * 128) return;
    int e = (int)(tid >> 7);
    int f = (int)(tid & 127) << 2;
    int r = row[e], c = col[e];
    float coef = dis[r] * w[e] * dis[c];
    const float* hp = h + (size_t)r * 512 + f;
    float* ap = acc + (size_t)c * 512 + f;
    atomicAdd(ap + 0, coef * hp[0]);
    atomicAdd(ap + 1, coef * hp[1]);
    atomicAdd(ap + 2, coef * hp[2]);
    atomicAdd(ap + 3, coef * hp[3]);
}

// AS[row,f] += w * P[col,f]   (F = 1024, 4 floats/thread)  -> A @ s
__global__ void scatter_as_kernel(const int* __restrict__ row, const int* __restrict__ col,
                                  const float* __restrict__ w,
                                  const float* __restrict__ P, float* __restrict__ AS, int E) {
    size_t tid = (size_t)blockIdx.x * blockDim.x + threadIdx.x;
    if (tid >= (size_t)E * 256) return;
    int e = (int)(tid >> 8);
    int f = (int)(tid & 255) << 2;
    int r = row[e], c = col[e];
    float we = w[e];
    const float* pp = P + (size_t)c * 1024 + f;
    float* ap = AS + (size_t)r * 1024 + f;
    atomicAdd(ap + 0, we * pp[0]);
    atomicAdd(ap + 1, we * pp[1]);
    atomicAdd(ap + 2, we * pp[2]);
    atomicAdd(ap + 3, we * pp[3]);
}

__global__ void bias_relu_kernel(float* __restrict__ x, const float* __restrict__ b,
                                 size_t n, int F) {
    size_t idx = (size_t)blockIdx.x * blockDim.x + threadIdx.x;
    if (idx < n) {
        int f = (int)(idx % F);
        x[idx] = fmaxf(x[idx] + b[f], 0.0f);
    }
}

__global__ void softmax_rows(const float* __restrict__ S, float* __restrict__ P, int cols) {
    __shared__ float red[256];
    const int row = blockIdx.x;
    const int tid = threadIdx.x;
    const float* s = S + (size_t)row * cols;
    float* p = P + (size_t)row * cols;
    float m = -INFINITY;
    for (int c = tid; c < cols; c += 256) m = fmaxf(m, s[c]);
    red[tid] = m; __syncthreads();
    for (int off = 128; off > 0; off >>= 1) {
        if (tid < off) red[tid] = fmaxf(red[tid], red[tid + off]);
        __syncthreads();
    }
    m = red[0]; __syncthreads();
    float sum = 0.0f;
    for (int c = tid; c < cols; c += 256) { float e = expf(s[c] - m); p[c] = e; sum += e; }
    red[tid] = sum; __syncthreads();
    for (int off = 128; off > 0; off >>= 1) {
        if (tid < off) red[tid] += red[tid + off];
        __syncthreads();
    }
    float inv = 1.0f / red[0];
    for (int c = tid; c < cols; c += 256) p[c] *= inv;
}

__global__ void trace_kernel(const float* __restrict__ Mx, int k, float* __restrict__ out) {
    __shared__ float red[256];
    float s = 0.0f;
    for (int i = threadIdx.x; i < k; i += 256) s += Mx[(size_t)i * k + i];
    red[threadIdx.x] = s; __syncthreads();
    for (int off = 128; off > 0; off >>= 1) {
        if (threadIdx.x < off) red[threadIdx.x] += red[threadIdx.x + off];
        __syncthreads();
    }
    if (threadIdx.x == 0) atomicAdd(out, red[0]);
}

// out += d[row] * sum_k P[row,k]^2     (one block per row)
__global__ void mincut_den_kernel(const float* __restrict__ P, const float* __restrict__ d,
                                  int cols, float* __restrict__ out) {
    __shared__ float red[256];
    const int row = blockIdx.x;
    const float* p = P + (size_t)row * cols;
    float s = 0.0f;
    for (int c = threadIdx.x; c < cols; c += 256) { float v = p[c]; s += v * v; }
    red[threadIdx.x] = s; __syncthreads();
    for (int off = 128; off > 0; off >>= 1) {
        if (threadIdx.x < off) red[threadIdx.x] += red[threadIdx.x + off];
        __syncthreads();
    }
    if (threadIdx.x == 0) atomicAdd(out, red[0] * d[row]);
}

__global__ void sumsq_kernel(const float* __restrict__ Mx, size_t n, float* __restrict__ out) {
    __shared__ float red[256];
    size_t idx = (size_t)blockIdx.x * 256 + threadIdx.x;
    float v = (idx < n) ? Mx[idx] : 0.0f;
    red[threadIdx.x] = v * v; __syncthreads();
    for (int off = 128; off > 0; off >>= 1) {
        if (threadIdx.x < off) red[threadIdx.x] += red[threadIdx.x + off];
        __syncthreads();
    }
    if (threadIdx.x == 0) atomicAdd(out, red[0]);
}

// out += sum (ss/||ss||_F - I/sqrt(k))^2
__global__ void ortho_kernel(const float* __restrict__ ss, int k,
                             const float* __restrict__ fro, float* __restrict__ out) {
    __shared__ float red[256];
    float nrm = sqrtf(*fro);
    float isq = rsqrtf((float)k);
    size_t n = (size_t)k * k;
    size_t idx = (size_t)blockIdx.x * 256 + threadIdx.x;
    float v = 0.0f;
    if (idx < n) {
        int i = (int)(idx / k), j = (int)(idx % k);
        float t = ss[idx] / nrm - ((i == j) ? isq : 0.0f);
        v = t * t;
    }
    red[threadIdx.x] = v; __syncthreads();
    for (int off = 128; off > 0; off >>= 1) {
        if (threadIdx.x < off) red[threadIdx.x] += red[threadIdx.x + off];
        __syncthreads();
    }
    if (threadIdx.x == 0) atomicAdd(out, red[0]);
}

// dd[row] = sqrt(sum_{c != row} adjraw[row,c]) + EPS
__global__ void dd_kernel(const float* __restrict__ adjraw, int k, float* __restrict__ dd) {
    __shared__ float red[256];
    const int row = blockIdx.x;
    float s = 0.0f;
    for (int c = threadIdx.x; c < k; c += 256)
        if (c != row) s += adjraw[(size_t)row * k + c];
    red[threadIdx.x] = s; __syncthreads();
    for (int off = 128; off > 0; off >>= 1) {
        if (threadIdx.x < off) red[threadIdx.x] += red[threadIdx.x + off];
        __syncthreads();
    }
    if (threadIdx.x == 0) dd[row] = sqrtf(red[0]) + EPSF;
}

__global__ void normadj_kernel(const float* __restrict__ adjraw, const float* __restrict__ dd,
                               float* __restrict__ adj, int k) {
    size_t n = (size_t)k * k;
    size_t idx = (size_t)blockIdx.x * blockDim.x + threadIdx.x;
    if (idx < n) {
        int i = (int)(idx / k), j = (int)(idx % k);
        adj[idx] = (i == j) ? 0.0f : adjraw[idx] / (dd[i] * dd[j]);
    }
}

__global__ void rowsum_kernel(const float* __restrict__ Mx, int k, float* __restrict__ out) {
    __shared__ float red[256];
    const int row = blockIdx.x;
    float s = 0.0f;
    for (int c = threadIdx.x; c < k; c += 256) s += Mx[(size_t)row * k + c];
    red[threadIdx.x] = s; __syncthreads();
    for (int off = 128; off > 0; off >>= 1) {
        if (threadIdx.x < off) red[threadIdx.x] += red[threadIdx.x + off];
        __syncthreads();
    }
    if (threadIdx.x == 0) out[row] = red[0];
}

// dis[i] = rsqrt(d[i] + 1)   (self-loop added degree of dense GCN)
__global__ void dis_selfloop_kernel(const float* __restrict__ d, float* __restrict__ dis, int n) {
    int i = blockIdx.x * blockDim.x + threadIdx.x;
    if (i < n) { float dg = d[i] + 1.0f; dis[i] = dg > 0.0f ? rsqrtf(dg) : 0.0f; }
}

// Ahat[i,j] = dis[i] * (adj[i,j] + (i==j)) * dis[j]
__global__ void ahat_kernel(const float* __restrict__ adj, const float* __restrict__ dis,
                            float* __restrict__ Ahat, int k) {
    size_t n = (size_t)k * k;
    size_t idx = (size_t)blockIdx.x * blockDim.x + threadIdx.x;
    if (idx < n) {
        int i = (int)(idx / k), j = (int)(idx % k);
        float a = adj[idx] + ((i == j) ? 1.0f : 0.0f);
        Ahat[idx] = dis[i] * a * dis[j];
    }
}

// out[row] = relu(dot(X[row,:], wv) + b0)   (N=1 linear layer)
__global__ void gemv_relu_kernel(const float* __restrict__ X, const float* __restrict__ wv,
                                 const float* __restrict__ b, float* __restrict__ out, int F) {
    __shared__ float red[256];
    const int row = blockIdx.x;
    float s = 0.0f;
    for (int c = threadIdx.x; c < F; c += 256) s += X[(size_t)row * F + c] * wv[c];
    red[threadIdx.x] = s; __syncthreads();
    for (int off = 128; off > 0; off >>= 1) {
        if (threadIdx.x < off) red[threadIdx.x] += red[threadIdx.x + off];
        __syncthreads();
    }
    if (threadIdx.x == 0) out[row] = fmaxf(red[0] + b[0], 0.0f);
}

// out[c] = sum_r X[r,c]    (stage-3 pooled features: softmax over 1 logit == 1)
__global__ void colsum_kernel(const float* __restrict__ X, int rows, int cols,
                              float* __restrict__ out) {
    int c = blockIdx.x * blockDim.x + threadIdx.x;
    if (c < cols) {
        float s = 0.0f;
        for (int r = 0; r < rows; ++r) s += X[(size_t)r * cols + c];
        out[c] = s;
    }
}

// stage-3 pooling (k=1) + loss writeout. 1 block, 128 threads.
__global__ void finalize_kernel(const float* __restrict__ adj2, const float* __restrict__ scal,
                                float* __restrict__ out_adj3, float* __restrict__ out_m,
                                float* __restrict__ out_o) {
    __shared__ float red[128];
    const int tid = threadIdx.x;
    float s = 0.0f;
    for (int c = 0; c < 128; ++c) s += adj2[(size_t)tid * 128 + c];
    red[tid] = s; __syncthreads();
    for (int off = 64; off > 0; off >>= 1) {
        if (tid < off) red[tid] += red[tid + off];
        __syncthreads();
    }
    if (tid == 0) {
        float total = red[0];              // trace(s^T A s) == den, s = ones
        float ss3 = 128.0f;                // s^T s (1x1), s_i = 1
        float nrm = fabsf(ss3);
        out_adj3[0] = 0.0f;                // diag-zeroed 1x1 adjacency
        out_m[0] = -(scal[0] / scal[1]);
        out_m[1] = -(scal[4] / scal[5]);
        out_m[2] = -(total / total);
        out_o[0] = sqrtf(scal[3]);
        out_o[1] = sqrtf(scal[7]);
        out_o[2] = fabsf(ss3 / nrm - 1.0f);
    }
}

// ---------------------------------------------------------------------------
extern "C" void kernel_launch(void* const* d_in, const int* in_sizes, int n_in,
                              void* d_out, int out_size, void* d_ws, size_t ws_size,
                              hipStream_t stream) {
    (void)in_sizes; (void)n_in; (void)out_size; (void)ws_size;

    const float* x   = (const float*)d_in[0];
    const int*   ei  = (const int*)d_in[1];
    const float* ew  = (const float*)d_in[2];
    const float* W1  = (const float*)d_in[3];
    const float* b1  = (const float*)d_in[4];
    const float* Wl1 = (const float*)d_in[5];
    const float* bl1 = (const float*)d_in[6];
    const float* W2  = (const float*)d_in[7];
    const float* b2  = (const float*)d_in[8];
    const float* Wl2 = (const float*)d_in[9];
    const float* bl2 = (const float*)d_in[10];
    const float* W3  = (const float*)d_in[11];
    const float* b3  = (const float*)d_in[12];
    const float* Wl3 = (const float*)d_in[13];
    const float* bl3 = (const float*)d_in[14];

    const int N1 = 10000, E = 320000, F = 512, C1 = 1024, C2 = 128;
    const int* erow = ei;
    const int* ecol = ei + E;

    float* ws = (float*)d_ws;
    size_t cursor = 0;
    auto take = [&](size_t n) { float* p = ws + cursor; cursor += n; return p; };

    float* h     = take((size_t)N1 * F);
    float* x1    = take((size_t)N1 * F);     // aggregation accumulator -> x1
    float* P1    = take((size_t)N1 * C1);
    float* AS    = take((size_t)N1 * C1);
    float* deg   = take(N1);
    float* dis   = take(N1);
    float* drow  = take(N1);
    float* x1b   = take((size_t)C1 * F);
    float* adj1r = take((size_t)C1 * C1);
    float* ss1   = take((size_t)C1 * C1);
    float* adj1  = take((size_t)C1 * C1);
    float* dd1   = take(C1);
    float* d2    = take(C1);
    float* dis2  = take(C1);
    float* Ahat2 = take((size_t)C1 * C1);
    float* t2    = take((size_t)C1 * F);
    float* x2    = take((size_t)C1 * F);
    float* P2    = take((size_t)C1 * C2);
    float* x2b   = take((size_t)C2 * F);
    float* AS2   = take((size_t)C1 * C2);
    float* adj2r = take((size_t)C2 * C2);
    float* ss2   = take((size_t)C2 * C2);
    float* adj2  = take((size_t)C2 * C2);
    float* dd2   = take(C2);
    float* d3    = take(C2);
    float* dis3  = take(C2);
    float* Ahat3 = take((size_t)C2 * C2);
    float* t3    = take((size_t)C2 * F);
    float* x3    = take((size_t)C2 * F);
    float* scal  = take(16);

    float* out      = (float*)d_out;
    float* out_x3b  = out;                           // 512
    float* out_adj3 = out + 512;                     // 1
    float* out_s1   = out + 513;                     // 10000*1024
    float* out_s2   = out_s1 + (size_t)N1 * C1;      // 1024*128
    float* out_s3   = out_s2 + (size_t)C1 * C2;      // 128
    float* out_m    = out_s3 + C2;                   // 3
    float* out_o    = out_m + 3;                     // 3

    const dim3 gblk(128);
    auto blks  = [](size_t n) { return (unsigned)((n + 255) / 256); };
    auto mgrid = [](int M) { return (unsigned)((M + 63) / 64); };

    // ---- init scratch scalars ----
    fill_kernel<<<1, 256, 0, stream>>>(scal, 0.0f, 16);

    // ================= Stage 1 (sparse GCN, n=10000) =================
    // h = x @ W1
    wmma_gemm<false><<<dim3(F / 64, mgrid(N1)), gblk, 0, stream>>>(x, W1, h, nullptr, N1, F, F, 0);

    fill_kernel<<<blks(N1), 256, 0, stream>>>(deg, 1.0f, N1);   // self-loop weight
    fill_kernel<<<blks(N1), 256, 0, stream>>>(drow, 0.0f, N1);
    edge_deg_kernel<<<blks(E), 256, 0, stream>>>(erow, ecol, ew, deg, drow, E);
    rsqrt_kernel<<<blks(N1), 256, 0, stream>>>(deg, dis, N1);

    selfloop_init_kernel<<<blks((size_t)N1 * F), 256, 0, stream>>>(h, dis, x1, (size_t)N1 * F, F);
    scatter_gcn_kernel<<<blks((size_t)E * 128), 256, 0, stream>>>(erow, ecol, ew, dis, h, x1, E);
    bias_relu_kernel<<<blks((size_t)N1 * F), 256, 0, stream>>>(x1, b1, (size_t)N1 * F, F);

    // s1 = relu(x1 @ Wl1 + bl1)  -> output
    wmma_gemm<false><<<dim3(C1 / 64, mgrid(N1)), gblk, 0, stream>>>(x1, Wl1, out_s1, bl1, N1, C1, F, 1);
    softmax_rows<<<N1, 256, 0, stream>>>(out_s1, P1, C1);

    // x1b = P1^T @ x1
    wmma_gemm<true><<<dim3(F / 64, mgrid(C1)), gblk, 0, stream>>>(P1, x1, x1b, nullptr, C1, F, N1, 0);

    // AS = A @ P1  (sparse scatter)
    fill_kernel<<<blks((size_t)N1 * C1), 256, 0, stream>>>(AS, 0.0f, (size_t)N1 * C1);
    scatter_as_kernel<<<blks((size_t)E * 256), 256, 0, stream>>>(erow, ecol, ew, P1, AS, E);

    // adj1_raw = P1^T @ AS ;  ss1 = P1^T @ P1
    wmma_gemm<true><<<dim3(C1 / 64, mgrid(C1)), gblk, 0, stream>>>(P1, AS, adj1r, nullptr, C1, C1, N1, 0);
    wmma_gemm<true><<<dim3(C1 / 64, mgrid(C1)), gblk, 0, stream>>>(P1, P1, ss1, nullptr, C1, C1, N1, 0);

    trace_kernel<<<1, 256, 0, stream>>>(adj1r, C1, scal + 0);
    mincut_den_kernel<<<N1, 256, 0, stream>>>(P1, drow, C1, scal + 1);
    sumsq_kernel<<<blks((size_t)C1 * C1), 256, 0, stream>>>(ss1, (size_t)C1 * C1, scal + 2);
    ortho_kernel<<<blks((size_t)C1 * C1), 256, 0, stream>>>(ss1, C1, scal + 2, scal + 3);
    dd_kernel<<<C1, 256, 0, stream>>>(adj1r, C1, dd1);
    normadj_kernel<<<blks((size_t)C1 * C1), 256, 0, stream>>>(adj1r, dd1, adj1, C1);

    // ================= Stage 2 (dense GCN, n=1024) =================
    rowsum_kernel<<<C1, 256, 0, stream>>>(adj1, C1, d2);
    dis_selfloop_kernel<<<blks(C1), 256, 0, stream>>>(d2, dis2, C1);
    ahat_kernel<<<blks((size_t)C1 * C1), 256, 0, stream>>>(adj1, dis2, Ahat2, C1);

    wmma_gemm<false><<<dim3(F / 64, mgrid(C1)), gblk, 0, stream>>>(x1b, W2, t2, nullptr, C1, F, F, 0);
    wmma_gemm<false><<<dim3(F / 64, mgrid(C1)), gblk, 0, stream>>>(Ahat2, t2, x2, b2, C1, F, C1, 1);
    wmma_gemm<false><<<dim3(C2 / 64, mgrid(C1)), gblk, 0, stream>>>(x2, Wl2, out_s2, bl2, C1, C2, F, 1);
    softmax_rows<<<C1, 256, 0, stream>>>(out_s2, P2, C2);

    wmma_gemm<true><<<dim3(F / 64, mgrid(C2)), gblk, 0, stream>>>(P2, x2, x2b, nullptr, C2, F, C1, 0);
    wmma_gemm<false><<<dim3(C2 / 64, mgrid(C1)), gblk, 0, stream>>>(adj1, P2, AS2, nullptr, C1, C2, C1, 0);
    wmma_gemm<true><<<dim3(C2 / 64, mgrid(C2)), gblk, 0, stream>>>(P2, AS2, adj2r, nullptr, C2, C2, C1, 0);
    wmma_gemm<true><<<dim3(C2 / 64, mgrid(C2)), gblk, 0, stream>>>(P2, P2, ss2, nullptr, C2, C2, C1, 0);

    trace_kernel<<<1, 256, 0, stream>>>(adj2r, C2, scal + 4);
    mincut_den_kernel<<<C1, 256, 0, stream>>>(P2, d2, C2, scal + 5);
    sumsq_kernel<<<blks((size_t)C2 * C2), 256, 0, stream>>>(ss2, (size_t)C2 * C2, scal + 6);
    ortho_kernel<<<blks((size_t)C2 * C2), 256, 0, stream>>>(ss2, C2, scal + 6, scal + 7);
    dd_kernel<<<C2, 256, 0, stream>>>(adj2r, C2, dd2);
    normadj_kernel<<<blks((size_t)C2 * C2), 256, 0, stream>>>(adj2r, dd2, adj2, C2);

    // ================= Stage 3 (dense GCN, n=128, k=1 pooling) =================
    rowsum_kernel<<<C2, 256, 0, stream>>>(adj2, C2, d3);
    dis_selfloop_kernel<<<1, 256, 0, stream>>>(d3, dis3, C2);
    ahat_kernel<<<blks((size_t)C2 * C2), 256, 0, stream>>>(adj2, dis3, Ahat3, C2);

    wmma_gemm<false><<<dim3(F / 64, mgrid(C2)), gblk, 0, stream>>>(x2b, W3, t3, nullptr, C2, F, F, 0);
    wmma_gemm<false><<<dim3(F / 64, mgrid(C2)), gblk, 0, stream>>>(Ahat3, t3, x3, b3, C2, F, C2, 1);

    gemv_relu_kernel<<<C2, 256, 0, stream>>>(x3, Wl3, bl3, out_s3, F);   // s3 output
    colsum_kernel<<<2, 256, 0, stream>>>(x3, C2, F, out_x3b);            // x3b = sum rows
    finalize_kernel<<<1, 128, 0, stream>>>(adj2, scal, out_adj3, out_m, out_o);
}